// My_net_22505628631731
// MI455X (gfx1250) — compile-verified
//
#include <hip/hip_runtime.h>
#include <hip/hip_bf16.h>
#include <math.h>

typedef __attribute__((ext_vector_type(2))) float v2f;
typedef __attribute__((ext_vector_type(8))) float v8f;

#define EPS 1e-7f
#define MD 10

__device__ inline float waveReduceAdd(float v) {
#pragma unroll
  for (int off = 16; off > 0; off >>= 1)
    v += __shfl_down(v, off, 32);
  return v;
}

// One block per (b, s) pair. Computes G[b, s, jidx, 0..9] = (sum_d g_m(z_d)/C_m(1))^2 / D
// and (for K==2, s<2) radius[b, ci*4 + q] = mean_d sigmoid(nrm*rad_w + rad_b).
template <int K, int KW>
__global__ __launch_bounds__(256) void branch_kernel(
    const float* __restrict__ x, const float* __restrict__ conv_w,
    const float* __restrict__ dir_b, const float* __restrict__ rad_w,
    const float* __restrict__ rad_b, float* __restrict__ G,
    float* __restrict__ radius, int jidx, int D, int L) {
  __shared__ float sw[K * KW];
  __shared__ float red[8][14];

  const int tid = threadIdx.x;
  const int b = blockIdx.x >> 3;
  const int s = blockIdx.x & 7;
  const int ci = s & 1;

  // Stage conv weights for this parity: w[i][t] = conv_w[(i*2+ci)*KW + t]
  if (tid < K * KW) {
    int i = tid / KW, t = tid % KW;
    sw[tid] = conv_w[(i * 2 + ci) * KW + t];
  }
  __syncthreads();

  // Normalized direction column s
  float nd[K];
  float dn = 0.f;
#pragma unroll
  for (int i = 0; i < K; ++i) {
    nd[i] = dir_b[i * 8 + s];
    dn = fmaf(nd[i], nd[i], dn);
  }
  float dinv = rsqrtf(dn);
#pragma unroll
  for (int i = 0; i < K; ++i) nd[i] *= dinv;

  // Precompute 1/C_m(1) for Gegenbauer normalization (p = K >= 3 path)
  float invc[MD];
  if constexpr (K != 2) {
    const float a = 0.5f * (float)(K - 2);
    float cm2 = 1.f, cm1 = 2.f * a;
    invc[0] = 1.f / cm1;
#pragma unroll
    for (int m = 2; m <= MD; ++m) {
      float c = (2.f * ((float)(m - 1) + a) * cm1 - ((float)(m - 2) + 2.f * a) * cm2) /
                (float)m;
      cm2 = cm1;
      cm1 = c;
      invc[m - 1] = 1.f / c;
    }
  }

  const float* xb = x + (size_t)b * L;
  float sum[MD];
#pragma unroll
  for (int m = 0; m < MD; ++m) sum[m] = 0.f;
  float rsum[4] = {0.f, 0.f, 0.f, 0.f};
  float rw[4] = {0.f, 0.f, 0.f, 0.f}, rb[4] = {0.f, 0.f, 0.f, 0.f};
  const bool doRad = (K == 2) && (s < 2);
  if (doRad) {
#pragma unroll
    for (int q = 0; q < 4; ++q) {
      rw[q] = rad_w[ci * 4 + q];
      rb[q] = rad_b[ci * 4 + q];
    }
  }

  for (int d = tid; d < D; d += 256) {
    float v[K];
#pragma unroll
    for (int i = 0; i < K; ++i) v[i] = 0.f;
#pragma unroll
    for (int t = 0; t < KW; ++t) {
      float xv = xb[d + t];
#pragma unroll
      for (int i = 0; i < K; ++i) v[i] = fmaf(sw[i * KW + t], xv, v[i]);
    }
    float n2 = 0.f;
#pragma unroll
    for (int i = 0; i < K; ++i) {
      v[i] += EPS;
      n2 = fmaf(v[i], v[i], n2);
    }
    float nrm = sqrtf(n2);
    float zd = 0.f;
#pragma unroll
    for (int i = 0; i < K; ++i) zd = fmaf(v[i], nd[i], zd);
    float z = zd / nrm;

    if constexpr (K == 2) {
      // Chebyshev T_m recurrence
      float gm2 = 1.f, gm1 = z;
      sum[0] += z;
#pragma unroll
      for (int m = 2; m <= MD; ++m) {
        float g = 2.f * z * gm1 - gm2;
        gm2 = gm1;
        gm1 = g;
        sum[m - 1] += g;
      }
    } else {
      const float a = 0.5f * (float)(K - 2);
      float gm2 = 1.f, gm1 = 2.f * a * z;
      sum[0] = fmaf(gm1, invc[0], sum[0]);
#pragma unroll
      for (int m = 2; m <= MD; ++m) {
        float g = (2.f * ((float)(m - 1) + a) * z * gm1 -
                   ((float)(m - 2) + 2.f * a) * gm2) *
                  (1.f / (float)m);
        gm2 = gm1;
        gm1 = g;
        sum[m - 1] = fmaf(g, invc[m - 1], sum[m - 1]);
      }
    }

    if (doRad) {
#pragma unroll
      for (int q = 0; q < 4; ++q) {
        float t2 = fmaf(nrm, rw[q], rb[q]);
        rsum[q] += 1.f / (1.f + expf(-t2));
      }
    }
  }

  // Block reduction (8 waves of 32)
  const int lane = tid & 31, wave = tid >> 5;
#pragma unroll
  for (int m = 0; m < MD; ++m) {
    float r = waveReduceAdd(sum[m]);
    if (lane == 0) red[wave][m] = r;
  }
  if (doRad) {
#pragma unroll
    for (int q = 0; q < 4; ++q) {
      float r = waveReduceAdd(rsum[q]);
      if (lane == 0) red[wave][MD + q] = r;
    }
  }
  __syncthreads();
  if (tid < MD) {
    float t = 0.f;
#pragma unroll
    for (int w = 0; w < 8; ++w) t += red[w][tid];
    G[(size_t)b * 240 + (size_t)(s * 3 + jidx) * MD + tid] = t * t / (float)D;
  }
  if (doRad && tid < 4) {
    float t = 0.f;
#pragma unroll
    for (int w = 0; w < 8; ++w) t += red[w][MD + tid];
    radius[b * 8 + ci * 4 + tid] = t / (float)D;
  }
}

// One block of 256 threads (8 waves): feats -> BN -> WMMA f32 MLP1 -> MLP2 -> log_softmax.
#define STR 60
__global__ __launch_bounds__(256) void head_kernel(
    const float* __restrict__ G, const float* __restrict__ radius,
    const float* __restrict__ weight_a, const float* __restrict__ bn_gamma,
    const float* __restrict__ bn_beta, const float* __restrict__ w1,
    const float* __restrict__ b1, const float* __restrict__ w2,
    const float* __restrict__ b2, float* __restrict__ out) {
  __shared__ float feats[64 * STR];  // (B=64, 56) padded rows (240B stride, 8B aligned)
  __shared__ float h[64 * 28];
  __shared__ float w1s[32 * 56];     // w1 zero-padded to 32 rows (N-tile padding)
  __shared__ float invn[48];
  const int tid = threadIdx.x;

  // Stage w1 into LDS, zero-padded rows 28..31 -> branch-free B-fragment loads later.
  for (int idx = tid; idx < 32 * 56; idx += 256)
    w1s[idx] = (idx < 28 * 56) ? w1[idx] : 0.0f;

  // inverse norms of weight_a over the degree axis, per (row, ch)
  if (tid < 48) {
    int row = tid >> 1, ch = tid & 1;
    float ss = 0.f;
#pragma unroll
    for (int m = 0; m < MD; ++m) {
      float w = weight_a[(row * MD + m) * 2 + ch];
      ss = fmaf(w, w, ss);
    }
    invn[tid] = rsqrtf(ss);
  }
  __syncthreads();

  // feats[:, 8:56] = sum_m G * wa  ;  feats[:, 0:8] = radius
  for (int idx = tid; idx < 64 * 48; idx += 256) {
    int bb = idx / 48, f = idx % 48;
    int row = f >> 1, ch = f & 1;
    const float* g = G + (size_t)bb * 240 + (size_t)row * MD;
    float acc = 0.f;
#pragma unroll
    for (int m = 0; m < MD; ++m) acc = fmaf(g[m], weight_a[(row * MD + m) * 2 + ch], acc);
    feats[bb * STR + 8 + f] = acc * invn[f];
  }
  for (int idx = tid; idx < 64 * 8; idx += 256) {
    int bb = idx >> 3, f = idx & 7;
    feats[bb * STR + f] = radius[idx];
  }
  __syncthreads();

  // Batch norm across the 64 rows (training-mode forward, population var)
  if (tid < 56) {
    float mu = 0.f;
    for (int bb = 0; bb < 64; ++bb) mu += feats[bb * STR + tid];
    mu *= (1.0f / 64.0f);
    float var = 0.f;
    for (int bb = 0; bb < 64; ++bb) {
      float d = feats[bb * STR + tid] - mu;
      var = fmaf(d, d, var);
    }
    var *= (1.0f / 64.0f);
    float sc = bn_gamma[tid] * rsqrtf(var + 1e-5f);
    float sh = bn_beta[tid];
    for (int bb = 0; bb < 64; ++bb)
      feats[bb * STR + tid] = (feats[bb * STR + tid] - mu) * sc + sh;
  }
  __syncthreads();

  // Layer 1 via V_WMMA_F32_16X16X4_F32: A = feats (64x56), B[k][n] = w1s[n][k].
  // 8 waves = 4 M-tiles x 2 N-tiles; 14 K-steps of 4. All fragments from LDS,
  // no divergence -> EXEC stays all-ones through the WMMA loop.
  const int lane = tid & 31, wave = tid >> 5;
  const int mt = wave >> 1, nt = wave & 1;
  const int mrow = lane & 15, half = lane >> 4;
  const int n = nt * 16 + mrow;
  const float* arow = feats + (mt * 16 + mrow) * STR + half * 2;
  const float* brow = w1s + n * 56 + half * 2;
  v8f acc = {};
#pragma unroll
  for (int ks = 0; ks < 56; ks += 4) {
    v2f a, bfrag;
    a.x = arow[ks];      // A frag: K = ks + 2*(lane/16) + vgpr
    a.y = arow[ks + 1];
    bfrag.x = brow[ks];  // B frag: same K indexing, N = lane%16 within tile
    bfrag.y = brow[ks + 1];
    acc = __builtin_amdgcn_wmma_f32_16x16x4_f32(
        /*neg_a=*/false, a, /*neg_b=*/false, bfrag,
        /*c_mod=*/(short)0, acc, /*reuse_a=*/false, /*reuse_b=*/false);
  }
  // D layout: VGPR r -> M = r + 8*(lane/16), N = lane%16
  if (n < 28) {
    float bias = b1[n];
#pragma unroll
    for (int r = 0; r < 8; ++r) {
      int M = mt * 16 + r + half * 8;
      float v = acc[r] + bias;
      h[M * 28 + n] = v > 0.f ? v : 0.f;
    }
  }
  __syncthreads();

  // Layer 2 + log_softmax
  if (tid < 64) {
    float o[4];
#pragma unroll
    for (int c = 0; c < 4; ++c) {
      float a2 = b2[c];
      for (int j = 0; j < 28; ++j) a2 = fmaf(h[tid * 28 + j], w2[c * 28 + j], a2);
      o[c] = a2;
    }
    float mx = fmaxf(fmaxf(o[0], o[1]), fmaxf(o[2], o[3]));
    float se = 0.f;
#pragma unroll
    for (int c = 0; c < 4; ++c) se += expf(o[c] - mx);
    float lse = mx + logf(se);
#pragma unroll
    for (int c = 0; c < 4; ++c) out[tid * 4 + c] = o[c] - lse;
  }
}

extern "C" void kernel_launch(void* const* d_in, const int* in_sizes, int n_in,
                              void* d_out, int out_size, void* d_ws, size_t ws_size,
                              hipStream_t stream) {
  (void)in_sizes; (void)n_in; (void)out_size; (void)ws_size;
  const float* x        = (const float*)d_in[0];   // (64,1,8192)
  const float* conv_w2  = (const float*)d_in[1];   // (4,1,3)
  const float* conv_w3  = (const float*)d_in[2];   // (6,1,4)
  const float* conv_w9  = (const float*)d_in[3];   // (18,1,10)
  const float* dir_b2   = (const float*)d_in[4];   // (2,8)
  const float* dir_b3   = (const float*)d_in[5];   // (3,8)
  const float* dir_b9   = (const float*)d_in[6];   // (9,8)
  const float* rad_w    = (const float*)d_in[7];   // (2,4)
  const float* rad_b    = (const float*)d_in[8];   // (2,4)
  const float* weight_a = (const float*)d_in[9];   // (24,10,2)
  const float* bn_gamma = (const float*)d_in[10];  // (56,)
  const float* bn_beta  = (const float*)d_in[11];  // (56,)
  const float* mlp_w1   = (const float*)d_in[12];  // (28,56)
  const float* mlp_b1   = (const float*)d_in[13];  // (28,)
  const float* mlp_w2   = (const float*)d_in[14];  // (4,28)
  const float* mlp_b2   = (const float*)d_in[15];  // (4,)
  float* out = (float*)d_out;                      // (64,4)

  float* G      = (float*)d_ws;       // 64*8*3*10 floats
  float* radius = G + 64 * 240;       // 64*8 floats

  const int L = 8192;
  dim3 blk(256);
  branch_kernel<2, 3><<<512, blk, 0, stream>>>(x, conv_w2, dir_b2, rad_w, rad_b,
                                               G, radius, 0, L - 3 + 1, L);
  branch_kernel<3, 4><<<512, blk, 0, stream>>>(x, conv_w3, dir_b3, rad_w, rad_b,
                                               G, radius, 1, L - 4 + 1, L);
  branch_kernel<9, 10><<<512, blk, 0, stream>>>(x, conv_w9, dir_b9, rad_w, rad_b,
                                                G, radius, 2, L - 10 + 1, L);
  head_kernel<<<1, blk, 0, stream>>>(G, radius, weight_a, bn_gamma, bn_beta,
                                     mlp_w1, mlp_b1, mlp_w2, mlp_b2, out);
}